// GATNet_29850022707668
// MI455X (gfx1250) — compile-verified
//
#include <hip/hip_runtime.h>
#include <hip/hip_bf16.h>
#include <math.h>

// GAT-2-layer for MI455X (gfx1250, wave32).
// GEMMs use V_WMMA_F32_16X16X4_F32 (full fp32 precision, CDNA5 matrix path).
// Edge softmax-aggregation done with L2-resident fp32 atomics (25.6MB accum fits 192MB L2).

typedef __attribute__((ext_vector_type(2))) float v2f;
typedef __attribute__((ext_vector_type(8))) float v8f;

#define IN_DIM 128
#define H1 8
#define HID 16
#define OUT_DIM 16
#define NEG_SLOPE 0.2f

__device__ __forceinline__ float lrelu(float x) { return x > 0.f ? x : NEG_SLOPE * x; }

// Monotone float atomic-max via integer atomics (valid for all-sign mixes, init -inf).
__device__ __forceinline__ void atomicMaxF(float* addr, float val) {
    if (val >= 0.f) atomicMax((int*)addr, __float_as_int(val));
    else            atomicMin((unsigned int*)addr, __float_as_uint(val));
}

__device__ __forceinline__ void edge_sd(int i, const int* __restrict__ es,
                                        const int* __restrict__ ed, int E,
                                        int& s, int& d) {
    if (i < E) { s = es[i]; d = ed[i]; } else { s = i - E; d = i - E; }
}

// ---------------- GEMM1: H1out[N,128] = X[N,128] @ W[128,128], fp32 WMMA ----------------
// One wave per 16-node tile; 8 column tiles of 16; K walked in steps of 4.
__global__ void __launch_bounds__(32) gemm1_wmma(const float* __restrict__ X,
                                                 const float* __restrict__ W,
                                                 float* __restrict__ Hout, int N) {
    int tile = blockIdx.x;
    int lane = threadIdx.x;
    int mrow = lane & 15;
    int g    = lane >> 4;              // lane group (0/1)
    int row  = tile * 16 + mrow;
    const float* xrow = X + (size_t)min(row, N - 1) * IN_DIM;

    v8f acc[8];
#pragma unroll
    for (int c = 0; c < 8; ++c) acc[c] = v8f{0.f, 0.f, 0.f, 0.f, 0.f, 0.f, 0.f, 0.f};

    for (int kk = 0; kk < IN_DIM / 4; ++kk) {
        int k0 = kk * 4 + g * 2;       // A VGPR v holds K = 2g+v  -> contiguous pair
        v2f a;
        a.x = xrow[k0];
        a.y = xrow[k0 + 1];
#pragma unroll
        for (int ct = 0; ct < 8; ++ct) {
            v2f b;                     // B VGPR v holds K = 2g+v, N = mrow
            b.x = W[(size_t)k0 * 128 + ct * 16 + mrow];
            b.y = W[(size_t)(k0 + 1) * 128 + ct * 16 + mrow];
            acc[ct] = __builtin_amdgcn_wmma_f32_16x16x4_f32(
                false, a, false, b, (short)0, acc[ct], false, false);
        }
    }
    // D: VGPR r -> M = r + 8*lanegroup. Fast path: full tile (always taken for N%16==0).
    if (tile * 16 + 16 <= N) {
#pragma unroll
        for (int ct = 0; ct < 8; ++ct) {
#pragma unroll
            for (int r = 0; r < 8; ++r) {
                int m = tile * 16 + r + 8 * g;
                Hout[(size_t)m * 128 + ct * 16 + mrow] = acc[ct][r];
            }
        }
    } else {
#pragma unroll
        for (int ct = 0; ct < 8; ++ct) {
#pragma unroll
            for (int r = 0; r < 8; ++r) {
                int m = tile * 16 + r + 8 * g;
                if (m < N) Hout[(size_t)m * 128 + ct * 16 + mrow] = acc[ct][r];
            }
        }
    }
}

// ---------------- GEMM2: H2[N,16] = H[N,128] @ W2[128,16] ----------------
__global__ void __launch_bounds__(32) gemm2_wmma(const float* __restrict__ H,
                                                 const float* __restrict__ W2,
                                                 float* __restrict__ H2, int N) {
    int tile = blockIdx.x;
    int lane = threadIdx.x;
    int mrow = lane & 15;
    int g    = lane >> 4;
    int row  = tile * 16 + mrow;
    const float* hrow = H + (size_t)min(row, N - 1) * 128;

    v8f acc = v8f{0.f, 0.f, 0.f, 0.f, 0.f, 0.f, 0.f, 0.f};
    for (int kk = 0; kk < 128 / 4; ++kk) {
        int k0 = kk * 4 + g * 2;
        v2f a; a.x = hrow[k0]; a.y = hrow[k0 + 1];
        v2f b;
        b.x = W2[(size_t)k0 * 16 + mrow];
        b.y = W2[(size_t)(k0 + 1) * 16 + mrow];
        acc = __builtin_amdgcn_wmma_f32_16x16x4_f32(
            false, a, false, b, (short)0, acc, false, false);
    }
    if (tile * 16 + 16 <= N) {
#pragma unroll
        for (int r = 0; r < 8; ++r) {
            int m = tile * 16 + r + 8 * g;
            H2[(size_t)m * 16 + mrow] = acc[r];
        }
    } else {
#pragma unroll
        for (int r = 0; r < 8; ++r) {
            int m = tile * 16 + r + 8 * g;
            if (m < N) H2[(size_t)m * 16 + mrow] = acc[r];
        }
    }
}

// ---------------- layer-1 attention logits + init ----------------
__global__ void s1_kernel(const float* __restrict__ h1,
                          const float* __restrict__ a_src, const float* __restrict__ a_dst,
                          float* __restrict__ ssrc, float* __restrict__ sdst,
                          float* __restrict__ m1, float* __restrict__ d1,
                          float* __restrict__ accum, int N) {
    int tid = blockIdx.x * blockDim.x + threadIdx.x;
    if (tid >= N * H1) return;
    int n = tid >> 3, h = tid & 7;
    const float* hp = h1 + (size_t)n * 128 + h * 16;
    float ss = 0.f, sd = 0.f;
#pragma unroll
    for (int c = 0; c < 16; ++c) {
        float v = hp[c];
        ss += v * a_src[h * 16 + c];
        sd += v * a_dst[h * 16 + c];
    }
    ssrc[tid] = ss; sdst[tid] = sd;
    m1[tid] = -INFINITY; d1[tid] = 0.f;
    float* z = accum + (size_t)n * 128 + h * 16;
#pragma unroll
    for (int c = 0; c < 16; ++c) z[c] = 0.f;
}

// ---------------- layer-1 edge passes ----------------
__global__ void passA1(const int* __restrict__ es, const int* __restrict__ ed, int E, int Etot,
                       const float* __restrict__ ssrc, const float* __restrict__ sdst,
                       float* __restrict__ m1) {
    int i = blockIdx.x * blockDim.x + threadIdx.x;
    if (i >= Etot) return;
    int s, d; edge_sd(i, es, ed, E, s, d);
#pragma unroll
    for (int h = 0; h < H1; ++h)
        atomicMaxF(&m1[d * H1 + h], lrelu(ssrc[s * H1 + h] + sdst[d * H1 + h]));
}

__global__ void passB1(const int* __restrict__ es, const int* __restrict__ ed, int E, int Etot,
                       const float* __restrict__ ssrc, const float* __restrict__ sdst,
                       const float* __restrict__ m1, float* __restrict__ d1) {
    int i = blockIdx.x * blockDim.x + threadIdx.x;
    if (i >= Etot) return;
    int s, d; edge_sd(i, es, ed, E, s, d);
#pragma unroll
    for (int h = 0; h < H1; ++h) {
        float e = lrelu(ssrc[s * H1 + h] + sdst[d * H1 + h]);
        atomicAdd(&d1[d * H1 + h], expf(e - m1[d * H1 + h]));
    }
}

// 32 lanes per edge: lane covers 4 features, head = lane/4.
__global__ void passC1(const int* __restrict__ es, const int* __restrict__ ed, int E, int Etot,
                       const float* __restrict__ ssrc, const float* __restrict__ sdst,
                       const float* __restrict__ m1, const float* __restrict__ d1,
                       const float* __restrict__ h1, float* __restrict__ accum) {
    int i = blockIdx.x * 8 + (threadIdx.x >> 5);
    if (i >= Etot) return;
    int lane = threadIdx.x & 31;
    int s, d; edge_sd(i, es, ed, E, s, d);
    int h = lane >> 2;
    float e = lrelu(ssrc[s * H1 + h] + sdst[d * H1 + h]);
    float alpha = expf(e - m1[d * H1 + h]) / d1[d * H1 + h];
    int f0 = lane * 4;
    const float4 v = *(const float4*)(h1 + (size_t)s * 128 + f0);
    float* o = accum + (size_t)d * 128 + f0;
    atomicAdd(o + 0, alpha * v.x);
    atomicAdd(o + 1, alpha * v.y);
    atomicAdd(o + 2, alpha * v.z);
    atomicAdd(o + 3, alpha * v.w);
}

// ---------------- ELU (+b1) in place ----------------
__global__ void elu_kernel(float* __restrict__ a, const float* __restrict__ b1, int total) {
    int i = blockIdx.x * blockDim.x + threadIdx.x;
    if (i >= total) return;
    float x = a[i] + b1[i & 127];
    a[i] = x > 0.f ? x : (expf(x) - 1.f);
}

// ---------------- layer-2 logits + init ----------------
__global__ void s2_kernel(const float* __restrict__ h2,
                          const float* __restrict__ a_src, const float* __restrict__ a_dst,
                          float* __restrict__ ssrc, float* __restrict__ sdst,
                          float* __restrict__ m2, float* __restrict__ d2,
                          float* __restrict__ out, int N) {
    int n = blockIdx.x * blockDim.x + threadIdx.x;
    if (n >= N) return;
    const float* hp = h2 + (size_t)n * 16;
    float ss = 0.f, sd = 0.f;
#pragma unroll
    for (int c = 0; c < 16; ++c) { ss += hp[c] * a_src[c]; sd += hp[c] * a_dst[c]; }
    ssrc[n] = ss; sdst[n] = sd;
    m2[n] = -INFINITY; d2[n] = 0.f;
    float* o = out + (size_t)n * 16;
#pragma unroll
    for (int c = 0; c < 16; ++c) o[c] = 0.f;
}

__global__ void passA2(const int* __restrict__ es, const int* __restrict__ ed, int E, int Etot,
                       const float* __restrict__ ssrc, const float* __restrict__ sdst,
                       float* __restrict__ m2) {
    int i = blockIdx.x * blockDim.x + threadIdx.x;
    if (i >= Etot) return;
    int s, d; edge_sd(i, es, ed, E, s, d);
    atomicMaxF(&m2[d], lrelu(ssrc[s] + sdst[d]));
}

__global__ void passB2(const int* __restrict__ es, const int* __restrict__ ed, int E, int Etot,
                       const float* __restrict__ ssrc, const float* __restrict__ sdst,
                       const float* __restrict__ m2, float* __restrict__ d2) {
    int i = blockIdx.x * blockDim.x + threadIdx.x;
    if (i >= Etot) return;
    int s, d; edge_sd(i, es, ed, E, s, d);
    float e = lrelu(ssrc[s] + sdst[d]);
    atomicAdd(&d2[d], expf(e - m2[d]));
}

// 16 lanes per edge.
__global__ void passC2(const int* __restrict__ es, const int* __restrict__ ed, int E, int Etot,
                       const float* __restrict__ ssrc, const float* __restrict__ sdst,
                       const float* __restrict__ m2, const float* __restrict__ d2,
                       const float* __restrict__ h2, float* __restrict__ out) {
    int i = blockIdx.x * 16 + (threadIdx.x >> 4);
    if (i >= Etot) return;
    int lane = threadIdx.x & 15;
    int s, d; edge_sd(i, es, ed, E, s, d);
    float e = lrelu(ssrc[s] + sdst[d]);
    float alpha = expf(e - m2[d]) / d2[d];
    atomicAdd(out + (size_t)d * 16 + lane, alpha * h2[(size_t)s * 16 + lane]);
}

// ---------------- final: +b2, log_softmax over 16, in place ----------------
__global__ void logsm_kernel(float* __restrict__ out, const float* __restrict__ b2, int N) {
    int n = blockIdx.x * blockDim.x + threadIdx.x;
    if (n >= N) return;
    float v[16];
    float* o = out + (size_t)n * 16;
    float m = -INFINITY;
#pragma unroll
    for (int c = 0; c < 16; ++c) { v[c] = o[c] + b2[c]; m = fmaxf(m, v[c]); }
    float ssum = 0.f;
#pragma unroll
    for (int c = 0; c < 16; ++c) ssum += expf(v[c] - m);
    float lse = m + logf(ssum);
#pragma unroll
    for (int c = 0; c < 16; ++c) o[c] = v[c] - lse;
}

extern "C" void kernel_launch(void* const* d_in, const int* in_sizes, int n_in,
                              void* d_out, int out_size, void* d_ws, size_t ws_size,
                              hipStream_t stream) {
    const float* x      = (const float*)d_in[0];
    const int*   eidx   = (const int*)d_in[1];
    const float* W1     = (const float*)d_in[2];
    const float* a_src1 = (const float*)d_in[3];
    const float* a_dst1 = (const float*)d_in[4];
    const float* b1     = (const float*)d_in[5];
    const float* W2     = (const float*)d_in[6];
    const float* a_src2 = (const float*)d_in[7];
    const float* a_dst2 = (const float*)d_in[8];
    const float* b2     = (const float*)d_in[9];

    const int N    = in_sizes[0] / IN_DIM;   // 50000
    const int E    = in_sizes[1] / 2;        // 800000
    const int Etot = E + N;                  // with self-loops
    const int* es = eidx;
    const int* ed = eidx + E;

    // workspace carve-up (~62 MB of fp32)
    float* ws = (float*)d_ws;
    float* h1    = ws; ws += (size_t)N * 128;
    float* accum = ws; ws += (size_t)N * 128;   // layer1 accum -> post-ELU features
    float* ssrc1 = ws; ws += (size_t)N * H1;
    float* sdst1 = ws; ws += (size_t)N * H1;
    float* m1    = ws; ws += (size_t)N * H1;
    float* d1    = ws; ws += (size_t)N * H1;
    float* h2    = ws; ws += (size_t)N * 16;
    float* ssrc2 = ws; ws += (size_t)N;
    float* sdst2 = ws; ws += (size_t)N;
    float* m2    = ws; ws += (size_t)N;
    float* d2    = ws; ws += (size_t)N;

    float* out = (float*)d_out;

    const int Ntiles = (N + 15) / 16;
    const int TB = 256;

    gemm1_wmma<<<Ntiles, 32, 0, stream>>>(x, W1, h1, N);
    s1_kernel<<<(N * H1 + TB - 1) / TB, TB, 0, stream>>>(h1, a_src1, a_dst1,
                                                         ssrc1, sdst1, m1, d1, accum, N);
    passA1<<<(Etot + TB - 1) / TB, TB, 0, stream>>>(es, ed, E, Etot, ssrc1, sdst1, m1);
    passB1<<<(Etot + TB - 1) / TB, TB, 0, stream>>>(es, ed, E, Etot, ssrc1, sdst1, m1, d1);
    passC1<<<(Etot + 7) / 8, TB, 0, stream>>>(es, ed, E, Etot, ssrc1, sdst1, m1, d1, h1, accum);
    elu_kernel<<<(N * 128 + TB - 1) / TB, TB, 0, stream>>>(accum, b1, N * 128);
    gemm2_wmma<<<Ntiles, 32, 0, stream>>>(accum, W2, h2, N);
    s2_kernel<<<(N + TB - 1) / TB, TB, 0, stream>>>(h2, a_src2, a_dst2,
                                                    ssrc2, sdst2, m2, d2, out, N);
    passA2<<<(Etot + TB - 1) / TB, TB, 0, stream>>>(es, ed, E, Etot, ssrc2, sdst2, m2);
    passB2<<<(Etot + TB - 1) / TB, TB, 0, stream>>>(es, ed, E, Etot, ssrc2, sdst2, m2, d2);
    passC2<<<(Etot + 15) / 16, TB, 0, stream>>>(es, ed, E, Etot, ssrc2, sdst2, m2, d2, h2, out);
    logsm_kernel<<<(N + TB - 1) / TB, TB, 0, stream>>>(out, b2, N);
}